// VE_Loss_71116068487811
// MI455X (gfx1250) — compile-verified
//
#include <hip/hip_runtime.h>
#include <math.h>

// ---------------------------------------------------------------------------
// VE_Loss for MI455X (gfx1250).
// pred: [8,3,256,256] f32, lab: [8,256,256] i32, out: 1 f32 scalar.
// Workspace layout (needs ~14.2 MB):
//   [0,        2 MB)  : pred_arg  int32 [8*65536]
//   [2 MB,    14 MB)  : dbuf      f32   [48*65536]  (f -> g^2 -> d, in place)
//   [14 MB, +256 B)   : chmax     f32   [48]
//   [ ... , +96 KB)   : partials  f32   [12*2048]
// ---------------------------------------------------------------------------

#define HH    256
#define WW    256
#define NB    8
#define NC    3
#define CHPIX (HH * WW)          // 65536
#define NPIX  (NB * CHPIX)       // 524288
#define NCH   (2 * NB * NC)      // 48
#define BIGF  1.0e6f
#define EPSF  1.0e-5f
#define NQ    12                 // reduced quantities per block in k_pixel

typedef __attribute__((ext_vector_type(2))) float v2f;
typedef __attribute__((ext_vector_type(8))) float v8f;

// ---------------- argmax over the 3 classes (first-max wins, like jnp) ------
__global__ void k_argmax(const float* __restrict__ pred, int* __restrict__ parg) {
    int p = blockIdx.x * 256 + threadIdx.x;
    if (p >= NPIX) return;
    int b = p / CHPIX, hw = p - b * CHPIX;
    const float* base = pred + (size_t)b * NC * CHPIX + hw;
    float x0 = base[0], x1 = base[CHPIX], x2 = base[2 * CHPIX];
    int a = 0; float m = x0;
    if (x1 > m) { m = x1; a = 1; }
    if (x2 > m) { m = x2; a = 2; }
    parg[p] = a;
}

// ---------------- EDT phase 1: per-column scans, writes g^2 -----------------
// blockIdx.x = channel t in [0,48), threadIdx.x = column w. Accesses are
// coalesced across the 256 lanes at each row h.
__global__ void k_colscan(const int* __restrict__ parg, const int* __restrict__ lab,
                          float* __restrict__ dbuf) {
    int t = blockIdx.x;
    int w = threadIdx.x;
    const int* src; int c;
    if (t < NB * NC) { src = parg + (size_t)(t / NC) * CHPIX; c = t % NC; }
    else             { int u = t - NB * NC; src = lab + (size_t)(u / NC) * CHPIX; c = u % NC; }
    float* out = dbuf + (size_t)t * CHPIX;

    float f = BIGF;                       // forward scan (matches jax.lax.scan init)
    for (int h = 0; h < HH; ++h) {
        f = (src[h * WW + w] == c) ? (f + 1.0f) : 0.0f;
        out[h * WW + w] = f;
    }
    float bb = BIGF;                      // backward scan, fuse min and square
    for (int h = HH - 1; h >= 0; --h) {
        bb = (src[h * WW + w] == c) ? (bb + 1.0f) : 0.0f;
        float g = fminf(out[h * WW + w], bb);
        out[h * WW + w] = g * g;
    }
}

// ---------------- EDT phase 2: brute-force min-plus per row -----------------
// One block per (channel,row). g2 row staged in LDS; lane j computes
// min_k (g2[k] + (j-k)^2) exactly as the reference (fma product is exact so
// rounding matches the two-step reference arithmetic), then sqrt, in place.
__global__ void k_minplus(float* __restrict__ dbuf) {
    __shared__ float sg[WW];
    int j = threadIdx.x;
    float* row = dbuf + (size_t)blockIdx.x * WW;
    sg[j] = row[j];
    __syncthreads();
    float fj = (float)j;
    float m = 3.402823466e+38f;
#pragma unroll 8
    for (int k = 0; k < WW; ++k) {
        float dk = fj - (float)k;
        m = fminf(m, fmaf(dk, dk, sg[k]));
    }
    row[j] = sqrtf(m);
}

// ---------------- per-channel max (deterministic block reduce) --------------
__global__ void k_chmax(const float* __restrict__ dbuf, float* __restrict__ chmax) {
    __shared__ float sm[256];
    int t = blockIdx.x, tid = threadIdx.x;
    const float* ch = dbuf + (size_t)t * CHPIX;
    float m = 0.0f;                       // distances are >= 0
    for (int i = tid; i < CHPIX; i += 256) m = fmaxf(m, ch[i]);
    sm[tid] = m; __syncthreads();
    for (int s = 128; s > 0; s >>= 1) {
        if (tid < s) sm[tid] = fmaxf(sm[tid], sm[tid + s]);
        __syncthreads();
    }
    if (tid == 0) chmax[t] = sm[0];
}

// ---------------- per-pixel loss terms + block partial sums -----------------
// Each block covers 256 contiguous pixels (within one batch image since
// 65536 % 256 == 0). 12 quantities reduced per block into partials[q][block].
__global__ void k_pixel(const float* __restrict__ pred, const int* __restrict__ lab,
                        const int* __restrict__ parg, const float* __restrict__ dbuf,
                        const float* __restrict__ chmax, float* __restrict__ part) {
    __shared__ float sm[NQ][256];
    int tid = threadIdx.x;
    int p = blockIdx.x * 256 + tid;
    int b = p / CHPIX, hw = p - b * CHPIX;

    // VE maps (normalized per channel, summed over classes, order c=0,1,2)
    float vp = 0.0f, vl = 0.0f;
    for (int c = 0; c < NC; ++c) {
        int tp = b * NC + c;
        int tl = NB * NC + b * NC + c;
        vp += dbuf[(size_t)tp * CHPIX + hw] / (chmax[tp] + EPSF);
        vl += dbuf[(size_t)tl * CHPIX + hw] / (chmax[tl] + EPSF);
    }
    int la = lab[p], pa = parg[p];
    // cosine over 4-channel features; one-hot dot = (la==pa), |onehot|^2 = 1
    float dot  = vp * vl + ((la == pa) ? 1.0f : 0.0f);
    float nl   = sqrtf(vp * vp + 1.0f);
    float npn  = sqrtf(vl * vl + 1.0f);
    float cosv = dot / fmaxf(nl * npn, 1e-8f);
    float msk  = (la > 0) ? 1.0f : 0.0f;

    // softmax / log-softmax (max-shifted, same as jax.nn)
    const float* base = pred + (size_t)b * NC * CHPIX + hw;
    float x0 = base[0], x1 = base[CHPIX], x2 = base[2 * CHPIX];
    float mx = fmaxf(x0, fmaxf(x1, x2));
    float e0 = expf(x0 - mx), e1 = expf(x1 - mx), e2 = expf(x2 - mx);
    float s  = e0 + e1 + e2;
    float lse = mx + logf(s);
    float xla = (la == 0) ? x0 : ((la == 1) ? x1 : x2);
    float logp = xla - lse;
    float inv = 1.0f / s;
    float p0 = e0 * inv, p1 = e1 * inv, p2 = e2 * inv;

    float q[NQ];
    q[0] = cosv * msk;                 q[1] = msk;                  q[2] = logp;
    q[3] = (la == 0) ? p0 : 0.0f;      q[4] = (la == 1) ? p1 : 0.0f; q[5] = (la == 2) ? p2 : 0.0f;
    q[6] = p0;                         q[7] = p1;                   q[8] = p2;
    q[9]  = (la == 0) ? 1.0f : 0.0f;   q[10] = (la == 1) ? 1.0f : 0.0f;
    q[11] = (la == 2) ? 1.0f : 0.0f;

    for (int i = 0; i < NQ; ++i) sm[i][tid] = q[i];
    __syncthreads();
    for (int s2 = 128; s2 > 0; s2 >>= 1) {
        if (tid < s2)
            for (int i = 0; i < NQ; ++i) sm[i][tid] += sm[i][tid + s2];
        __syncthreads();
    }
    if (tid < NQ) part[tid * 2048 + blockIdx.x] = sm[tid][0];
}

// ---------------- final: single-wave WMMA ones-reduction + combine ----------
// D = A x B(=ones) + C accumulates per-row sums of 64 values per issue in
// full f32 via V_WMMA_F32_16X16X4_F32. EXEC is all ones (block of exactly 32,
// no divergence). Fixed reduction order -> deterministic output.
__device__ inline float wave_wmma_sum(const float* __restrict__ arr, int n) {
    int lane = threadIdx.x;              // block is exactly one wave32
    v2f ones; ones.x = 1.0f; ones.y = 1.0f;
    v8f c = {0.f, 0.f, 0.f, 0.f, 0.f, 0.f, 0.f, 0.f};
    for (int i = 0; i < n; i += 64) {
        v2f a;
        a.x = arr[i + lane];
        a.y = arr[i + 32 + lane];
        c = __builtin_amdgcn_wmma_f32_16x16x4_f32(
                /*neg_a=*/false, a, /*neg_b=*/false, ones,
                /*c_mod=*/(short)0, c, /*reuse_a=*/false, /*reuse_b=*/false);
    }
    // D[m][0]: m=0..7 live in lane 0 (vgpr m), m=8..15 in lane 16 (vgpr m-8).
    float partial = c[0] + c[1] + c[2] + c[3] + c[4] + c[5] + c[6] + c[7];
    return __shfl(partial, 0, 32) + __shfl(partial, 16, 32);
}

__global__ void k_final(const float* __restrict__ part, float* __restrict__ out) {
    const float* cosP  = part;
    const float* maskP = part + 2048;
    const float* logpP = part + 2 * 2048;

    float loss_ve = 0.0f;
    for (int b = 0; b < NB; ++b) {     // per-batch masked cosine mean
        float sc = wave_wmma_sum(cosP  + b * 256, 256);
        float sk = wave_wmma_sum(maskP + b * 256, 256);
        loss_ve += 1.0f - sc / fmaxf(sk, 1.0f);
    }
    float ce = -wave_wmma_sum(logpP, 2048) / (float)NPIX;

    float dsum = 0.0f;
    for (int c = 0; c < NC; ++c) {
        float inter = wave_wmma_sum(part + (3 + c) * 2048, 2048);
        float sump  = wave_wmma_sum(part + (6 + c) * 2048, 2048);
        float slab  = wave_wmma_sum(part + (9 + c) * 2048, 2048);
        dsum += (2.0f * inter + EPSF) / (sump + slab + EPSF);
    }
    float dice_loss = 1.0f - dsum / 3.0f;

    if (threadIdx.x == 0)
        out[0] = 1.0f * ce + 0.6f * dice_loss + 0.5f * loss_ve;
}

// ---------------------------------------------------------------------------
extern "C" void kernel_launch(void* const* d_in, const int* in_sizes, int n_in,
                              void* d_out, int out_size, void* d_ws, size_t ws_size,
                              hipStream_t stream) {
    const float* pred = (const float*)d_in[0];
    const int*   lab  = (const int*)d_in[1];

    char*  ws    = (char*)d_ws;
    int*   parg  = (int*)(ws);                                   // 2 MB
    float* dbuf  = (float*)(ws + 2097152);                       // 12 MB
    float* chmax = (float*)(ws + 2097152 + 12582912);            // 192 B (pad 256)
    float* part  = (float*)(ws + 2097152 + 12582912 + 256);      // 96 KB

    k_argmax <<<NPIX / 256, 256, 0, stream>>>(pred, parg);
    k_colscan<<<NCH,        256, 0, stream>>>(parg, lab, dbuf);
    k_minplus<<<NCH * HH,   256, 0, stream>>>(dbuf);
    k_chmax  <<<NCH,        256, 0, stream>>>(dbuf, chmax);
    k_pixel  <<<NPIX / 256, 256, 0, stream>>>(pred, lab, parg, dbuf, chmax, part);
    k_final  <<<1,           32, 0, stream>>>(part, (float*)d_out);
}